// ResidualBlock3_62551903699474
// MI455X (gfx1250) — compile-verified
//
#include <hip/hip_runtime.h>
#include <hip/hip_bf16.h>

#define DFEAT 96
#define TILE 16
#define COLT (DFEAT / TILE)   // 6 column tiles
#define BN_EPS 1e-3f

typedef float v2f __attribute__((ext_vector_type(2)));
typedef float v8f __attribute__((ext_vector_type(8)));

// ---------------------------------------------------------------------------
// H = X @ W using V_WMMA_F32_16X16X4_F32 (full f32 precision, matrix pipe).
// One wave32 computes one 16x16 tile of H; K=96 consumed in 24 steps of 4.
// W (96x96 = 36KB) staged in LDS.
// ---------------------------------------------------------------------------
__global__ __launch_bounds__(256) void gemm_wmma_kernel(
    const float* __restrict__ X, const float* __restrict__ W,
    float* __restrict__ H, int nrows)
{
    __shared__ float sW[DFEAT * DFEAT];
    {
        const float4* __restrict__ Wv = (const float4*)W;
        float4* sWv = (float4*)sW;
        for (int i = threadIdx.x; i < (DFEAT * DFEAT) / 4; i += blockDim.x)
            sWv[i] = Wv[i];
    }
    __syncthreads();

    const int lane = threadIdx.x & 31;
    const int wave = (int)((blockIdx.x * blockDim.x + threadIdx.x) >> 5);
    const int rowTile = wave / COLT;
    const int colTile = wave - rowTile * COLT;
    if (rowTile * TILE >= nrows) return;   // wave-uniform exit; EXEC stays full

    const int m  = lane & 15;
    const int kh = (lane >> 4) << 1;           // 0 for lanes 0-15, 2 for 16-31
    const int n  = colTile * TILE + m;

    const float* __restrict__ arow = X + (size_t)(rowTile * TILE + m) * DFEAT + kh;
    const float* __restrict__ bcol = sW + kh * DFEAT + n;

    v8f c = {};
#pragma unroll
    for (int k0 = 0; k0 < DFEAT; k0 += 4) {
        // A fragment: lane m holds X[row=m][k0+kh], X[row=m][k0+kh+1]
        v2f a = *(const v2f*)(arow + k0);
        // B fragment: lane holds W[k0+kh][n], W[k0+kh+1][n] from LDS
        v2f b;
        b.x = bcol[k0 * DFEAT];
        b.y = bcol[k0 * DFEAT + DFEAT];
        c = __builtin_amdgcn_wmma_f32_16x16x4_f32(
                /*neg_a=*/false, a, /*neg_b=*/false, b,
                /*c_mod=*/(short)0, c, /*reuse_a=*/false, /*reuse_b=*/false);
    }

    // C/D layout: VGPR v -> lanes 0-15: M=v, N=lane; lanes 16-31: M=v+8.
    const int mbase = rowTile * TILE + ((lane >> 4) << 3);
#pragma unroll
    for (int v = 0; v < 8; ++v) {
        int r = mbase + v;
        if (r < nrows) H[(size_t)r * DFEAT + n] = c[v];
    }
}

// ---------------------------------------------------------------------------
// Zero the aggregation buffer.
// ---------------------------------------------------------------------------
__global__ void zero_kernel(float* __restrict__ p, int n)
{
    int i = blockIdx.x * blockDim.x + threadIdx.x;
    if (i < n) p[i] = 0.0f;
}

// ---------------------------------------------------------------------------
// Scatter: agg[dst] += edge_w * H[src].  One wave32 per edge; 96 features in
// 3 coalesced chunks of 32 -> 3 global_atomic_add_f32 per lane. L2 resident.
// ---------------------------------------------------------------------------
__global__ __launch_bounds__(256) void scatter_kernel(
    const float* __restrict__ H,
    const int* __restrict__ esrc, const int* __restrict__ edst,
    const float* __restrict__ ew,
    float* __restrict__ agg, int nedges)
{
    const int wave = (int)((blockIdx.x * blockDim.x + threadIdx.x) >> 5);
    const int lane = threadIdx.x & 31;
    if (wave >= nedges) return;

    const int   s = esrc[wave];
    const int   d = edst[wave];
    const float w = ew[wave];
    const float* __restrict__ hs = H   + (size_t)s * DFEAT;
    float*       __restrict__ ad = agg + (size_t)d * DFEAT;

#pragma unroll
    for (int i = 0; i < 3; ++i) {
        int col = lane + 32 * i;
        atomicAdd(ad + col, w * hs[col]);
    }
}

// ---------------------------------------------------------------------------
// out = bn(relu(agg)) [+ residual].  Inference BN, eps = 1e-3.
// ---------------------------------------------------------------------------
__global__ void bn_relu_kernel(
    const float* __restrict__ agg,
    const float* __restrict__ g, const float* __restrict__ b,
    const float* __restrict__ mean, const float* __restrict__ var,
    const float* __restrict__ residual,   // nullptr if none
    float* __restrict__ out, int total)
{
    int i = blockIdx.x * blockDim.x + threadIdx.x;
    if (i >= total) return;
    int col = i % DFEAT;
    float x = fmaxf(agg[i], 0.0f);
    float scale = g[col] * rsqrtf(var[col] + BN_EPS);
    float y = (x - mean[col]) * scale + b[col];
    if (residual) y += residual[i];
    out[i] = y;
}

// ---------------------------------------------------------------------------
// Orchestration: 3 layers of (GEMM -> zero -> scatter -> bn/relu).
// d_out doubles as the ping buffer for the layer inputs (written by bn_relu,
// read by the next gemm). Workspace: Htmp + Agg = 2 * N * 96 floats.
// ---------------------------------------------------------------------------
extern "C" void kernel_launch(void* const* d_in, const int* in_sizes, int n_in,
                              void* d_out, int out_size, void* d_ws, size_t ws_size,
                              hipStream_t stream)
{
    const float* x    = (const float*)d_in[0];
    const int*   esrc = (const int*)  d_in[1];
    const int*   edst = (const int*)  d_in[2];
    const float* ew   = (const float*)d_in[3];
    const float* W[3] = { (const float*)d_in[4], (const float*)d_in[5], (const float*)d_in[6] };
    // setup order: g1,b1,m1,v1, g2,b2,m2,v2, g3,b3,m3,v3 at indices 7..18
    const float* G[3] = { (const float*)d_in[7],  (const float*)d_in[11], (const float*)d_in[15] };
    const float* B[3] = { (const float*)d_in[8],  (const float*)d_in[12], (const float*)d_in[16] };
    const float* M[3] = { (const float*)d_in[9],  (const float*)d_in[13], (const float*)d_in[17] };
    const float* V[3] = { (const float*)d_in[10], (const float*)d_in[14], (const float*)d_in[18] };

    const int N = in_sizes[0] / DFEAT;
    const int E = in_sizes[1];
    const int total = N * DFEAT;

    float* out  = (float*)d_out;
    float* Htmp = (float*)d_ws;
    float* Agg  = Htmp + (size_t)total;

    const int tiles      = ((N + TILE - 1) / TILE) * COLT;
    const int gemmBlocks = (tiles + 7) / 8;         // 8 waves / 256-thread block
    const int ptBlocks   = (total + 255) / 256;
    const int scatBlocks = (E + 7) / 8;             // one wave per edge

    const float* cur = x;
    for (int l = 0; l < 3; ++l) {
        gemm_wmma_kernel<<<gemmBlocks, 256, 0, stream>>>(cur, W[l], Htmp, N);
        zero_kernel<<<ptBlocks, 256, 0, stream>>>(Agg, total);
        scatter_kernel<<<scatBlocks, 256, 0, stream>>>(Htmp, esrc, edst, ew, Agg, E);
        bn_relu_kernel<<<ptBlocks, 256, 0, stream>>>(
            Agg, G[l], B[l], M[l], V[l], (l == 2) ? x : nullptr, out, total);
        cur = out;
    }
}